// SpectralFMPretrainLoss_56719338111043
// MI455X (gfx1250) — compile-verified
//
#include <hip/hip_runtime.h>
#include <hip/hip_bf16.h>
#include <math.h>

typedef __attribute__((ext_vector_type(2))) float v2f;
typedef __attribute__((ext_vector_type(8))) float v8f;
typedef __attribute__((ext_vector_type(4))) unsigned int u32x4;
typedef __attribute__((ext_vector_type(8))) unsigned int u32x8;

#define NN 4096
#define DD 512
#define FEPS 1e-8f

// -------- workspace layout (in floats) --------
// acc[0]=msrp_sq acc[1]=msrp_cnt acc[2]=contrast_sum acc[3]=ot_sum
// acc[4]=smooth  acc[5]=nonneg   acc[6]=deriv        acc[7]=peak_num acc[8]=peak_cnt
#define ACC_OFF   0
#define XSQ_OFF   64
#define YSQ_OFF   (XSQ_OFF + NN)
#define U_OFF     (YSQ_OFF + NN)
#define V_OFF     (U_OFF + NN)
#define PART_OFF  (V_OFF + NN)          // 16*4096 floats of split-K partials
#define S_OFF     131072                // Gram matrix S = z1 @ z2^T (64 MB)
#define K_OFF     (S_OFF + NN * NN)     // Sinkhorn kernel K = exp(-C/reg) (64 MB)

// ---------------- wave helpers (wave32) ----------------
__device__ __forceinline__ float wave_sum(float x) {
#pragma unroll
  for (int o = 16; o > 0; o >>= 1) x += __shfl_xor(x, o, 32);
  return x;
}
__device__ __forceinline__ float wave_max(float x) {
#pragma unroll
  for (int o = 16; o > 0; o >>= 1) x = fmaxf(x, __shfl_xor(x, o, 32));
  return x;
}

// ---------------- Tensor Data Mover: 2D tile load (global -> LDS) ----------------
// D# groups 0/1 only (<=2D tensor). data_size = 4B. Padding: 1 DWORD inserted every
// 4 DWORDs -> a 16-DWORD tile row occupies a 20-DWORD LDS stride (bank-conflict-free
// WMMA fragment reads: banks 20*m mod 64 are distinct over m=0..15).
__device__ __forceinline__ void tdm_load_2d(const float* gaddr, unsigned lds_byte_off,
                                            unsigned tile_rows) {
  unsigned long long ga = (unsigned long long)(size_t)gaddr;
  u32x4 g0;
  g0[0] = 1u;                                            // count=1 (valid descriptor)
  g0[1] = lds_byte_off;                                  // lds_addr (bytes)
  g0[2] = (unsigned)(ga & 0xffffffffu);                  // global_addr[31:0]
  g0[3] = (unsigned)((ga >> 32) & 0x01ffffffu) | (2u << 30); // global_addr[56:32] | type=2
  u32x8 g1;
  g1[0] = 0x00520000u;        // data_size=4B(2)<<16 | pad_enable(b20) | pad_interval=4dw(1)<<22 | pad_amount=1dw(0)
  g1[1] = (unsigned)DD << 16; // tensor_dim0 = 512  (bits 79:48, low half here)
  g1[2] = (unsigned)NN << 16; // tensor_dim1 = 4096 (bits 111:80, low half here)
  g1[3] = 16u << 16;          // tile_dim0 = 16 (bits 127:112)
  g1[4] = tile_rows;          // tile_dim1 (bits 143:128)
  g1[5] = (unsigned)DD;       // tensor_dim0_stride = 512 (bits 207:160, low 32)
  g1[6] = 0u;
  g1[7] = 0u;
  asm volatile("tensor_load_to_lds %0, %1" :: "s"(g0), "s"(g1) : "memory");
}

// ---------------- init: zero accumulators, u = v = 1 ----------------
__global__ void k_init(float* __restrict__ w) {
  int t = blockIdx.x * blockDim.x + threadIdx.x;
  if (t < 64) w[ACC_OFF + t] = 0.0f;
  if (t < NN) { w[U_OFF + t] = 1.0f; w[V_OFF + t] = 1.0f; }
}

// ---------------- row squared norms of z1 (rows 0..4095) and z2 (4096..8191) ----------------
__global__ void __launch_bounds__(256) k_rownorms(const float* __restrict__ z1,
                                                  const float* __restrict__ z2,
                                                  float* __restrict__ w) {
  int wave = threadIdx.x >> 5, lane = threadIdx.x & 31;
  int row = blockIdx.x * 8 + wave;            // 0..8191
  const float* src = (row < NN) ? (z1 + (size_t)row * DD) : (z2 + (size_t)(row - NN) * DD);
  float acc = 0.0f;
#pragma unroll
  for (int it = 0; it < DD / 32; ++it) {
    float a = src[lane + it * 32];
    acc += a * a;
  }
  acc = wave_sum(acc);
  if (lane == 0) {
    if (row < NN) w[XSQ_OFF + row] = acc;
    else          w[YSQ_OFF + row - NN] = acc;
  }
}

// ---------------- S = z1 @ z2^T via V_WMMA_F32_16X16X4_F32 ----------------
// block = 256 thr (8 wave32), macro-tile 128(M) x 64(N).
// K staged 16 at a time by the Tensor Data Mover into double-buffered LDS,
// DMA of chunk c+1 overlapped with the 16 WMMAs of chunk c.
__global__ void __launch_bounds__(256) k_gemm(const float* __restrict__ z1,
                                              const float* __restrict__ z2,
                                              float* __restrict__ S) {
  // A tile: 128 rows x 20 dw (padded) = 2560 dw; B tile: 64 x 20 = 1280 dw. x2 buffers.
  __shared__ float ls[2 * 2560 + 2 * 1280];
  const int m0 = blockIdx.x * 128;
  const int n0 = blockIdx.y * 64;
  const int tid  = threadIdx.x;
  const int lane = tid & 31;
  const int wave = tid >> 5;
  const int half = lane >> 4;     // ISA A/B frag: lanes>=16 carry K+2
  const int l16  = lane & 15;
  const int kA   = 2 * half;      // VGPR0 holds K = kA, VGPR1 holds K = kA+1

  const unsigned ldsBase = (unsigned)(size_t)(&ls[0]);   // LDS byte offset of buffer
  const unsigned A_OF0 = 0u,     A_OF1 = 2560u;          // float offsets
  const unsigned B_OF0 = 5120u,  B_OF1 = 6400u;

  v8f acc0 = {}, acc1 = {}, acc2 = {}, acc3 = {};

  if (wave == 0) {
    tdm_load_2d(z1 + (size_t)m0 * DD, ldsBase + A_OF0 * 4u, 128u);
    tdm_load_2d(z2 + (size_t)n0 * DD, ldsBase + B_OF0 * 4u, 64u);
  }

  for (int c = 0; c < DD / 16; ++c) {
    if (wave == 0) __builtin_amdgcn_s_wait_tensorcnt(0); // chunk c landed in LDS
    __syncthreads();                                     // publish to all waves; buffer c+1 free
    if (wave == 0 && c + 1 < DD / 16) {                  // kick DMA for chunk c+1 (other buffer)
      const int k0n = (c + 1) * 16;
      const unsigned aof = ((c + 1) & 1) ? A_OF1 : A_OF0;
      const unsigned bof = ((c + 1) & 1) ? B_OF1 : B_OF0;
      tdm_load_2d(z1 + (size_t)m0 * DD + k0n, ldsBase + aof * 4u, 128u);
      tdm_load_2d(z2 + (size_t)n0 * DD + k0n, ldsBase + bof * 4u, 64u);
    }
    const float* As = ls + ((c & 1) ? A_OF1 : A_OF0);
    const float* Bs = ls + ((c & 1) ? B_OF1 : B_OF0);
#pragma unroll
    for (int kk = 0; kk < 16; kk += 4) {
      const int k    = kk + kA;
      const int kidx = k + (k >> 2);            // padded-row index; k even -> k+1 adjacent
      const int ml   = wave * 16 + l16;
      v2f a;
      a.x = As[ml * 20 + kidx];
      a.y = As[ml * 20 + kidx + 1];
      v2f b0, b1, b2, b3;
      b0.x = Bs[l16 * 20 + kidx];        b0.y = Bs[l16 * 20 + kidx + 1];
      b1.x = Bs[(16 + l16) * 20 + kidx]; b1.y = Bs[(16 + l16) * 20 + kidx + 1];
      b2.x = Bs[(32 + l16) * 20 + kidx]; b2.y = Bs[(32 + l16) * 20 + kidx + 1];
      b3.x = Bs[(48 + l16) * 20 + kidx]; b3.y = Bs[(48 + l16) * 20 + kidx + 1];
      acc0 = __builtin_amdgcn_wmma_f32_16x16x4_f32(false, a, false, b0, (short)0, acc0, false, false);
      acc1 = __builtin_amdgcn_wmma_f32_16x16x4_f32(false, a, false, b1, (short)0, acc1, false, false);
      acc2 = __builtin_amdgcn_wmma_f32_16x16x4_f32(false, a, false, b2, (short)0, acc2, false, false);
      acc3 = __builtin_amdgcn_wmma_f32_16x16x4_f32(false, a, false, b3, (short)0, acc3, false, false);
    }
    __syncthreads();                                     // done reading buffer c
  }

  // C/D layout: VGPR r -> M = r (lanes 0-15) or r+8 (lanes 16-31), N = lane&15
#pragma unroll
  for (int r = 0; r < 8; ++r) {
    int row = m0 + wave * 16 + r + half * 8;
    size_t base = (size_t)row * NN + n0 + l16;
    S[base]      = acc0[r];
    S[base + 16] = acc1[r];
    S[base + 32] = acc2[r];
    S[base + 48] = acc3[r];
  }
}

// ---------------- per-row: contrastive log-softmax diag + K = exp(-C/reg) ----------------
__global__ void __launch_bounds__(256) k_rows(float* __restrict__ w) {
  __shared__ float sh[8];
  const float* S  = w + S_OFF;
  float*       Km = w + K_OFF;
  const int i = blockIdx.x;
  const int tid = threadIdx.x;
  const float xs    = w[XSQ_OFF + i];
  const float inv_i = rsqrtf(xs);

  // pass 1: row max of logits; also emit K row
  float lmax = -3.402823466e38f;
  for (int j = tid; j < NN; j += 256) {
    float s  = S[(size_t)i * NN + j];
    float ys = w[YSQ_OFF + j];
    float l  = s * inv_i * rsqrtf(ys) * 10.0f;            // 1/TEMP
    lmax = fmaxf(lmax, l);
    float c = fmaxf(xs + ys - 2.0f * s, 0.0f);
    Km[(size_t)i * NN + j] = __expf(-20.0f * c);           // 1/REG
  }
  lmax = wave_max(lmax);
  if ((tid & 31) == 0) sh[tid >> 5] = lmax;
  __syncthreads();
  float rowmax = sh[0];
#pragma unroll
  for (int t = 1; t < 8; ++t) rowmax = fmaxf(rowmax, sh[t]);
  __syncthreads();

  // pass 2: sum exp(l - max)
  float lsum = 0.0f;
  for (int j = tid; j < NN; j += 256) {
    float s  = S[(size_t)i * NN + j];
    float l  = s * inv_i * rsqrtf(w[YSQ_OFF + j]) * 10.0f;
    lsum += __expf(l - rowmax);
  }
  lsum = wave_sum(lsum);
  if ((tid & 31) == 0) sh[tid >> 5] = lsum;
  __syncthreads();
  if (tid == 0) {
    float tot = 0.0f;
#pragma unroll
    for (int t = 0; t < 8; ++t) tot += sh[t];
    float lse  = rowmax + __logf(tot);
    float l_ii = S[(size_t)i * NN + i] * inv_i * rsqrtf(w[YSQ_OFF + i]) * 10.0f;
    atomicAdd(w + ACC_OFF + 2, -(l_ii - lse));             // sum of -logp diag
  }
}

// ---------------- Sinkhorn: v = nu / (K^T u + eps) ----------------
// split-K partials: grid (16 j-blocks, 16 i-chunks), thread-per-column (coalesced)
__global__ void __launch_bounds__(256) k_vpart(const float* __restrict__ w, float* __restrict__ part) {
  const float* Km = w + K_OFF;
  const float* u  = w + U_OFF;
  const int j  = blockIdx.x * 256 + threadIdx.x;
  const int i0 = blockIdx.y * 256;
  float acc = 0.0f;
#pragma unroll 4
  for (int ii = 0; ii < 256; ++ii)
    acc += Km[(size_t)(i0 + ii) * NN + j] * u[i0 + ii];
  part[blockIdx.y * NN + j] = acc;
}
__global__ void k_vfin(float* __restrict__ w) {
  const float* part = w + PART_OFF;
  int j = blockIdx.x * blockDim.x + threadIdx.x;
  float s = 0.0f;
#pragma unroll
  for (int c = 0; c < 16; ++c) s += part[c * NN + j];
  w[V_OFF + j] = (1.0f / NN) / (s + FEPS);
}

// ---------------- Sinkhorn: u = mu / (K v + eps) ----------------
// wave-per-row, lane-strided columns (coalesced), 4 column chunks
__global__ void __launch_bounds__(256) k_upart(const float* __restrict__ w, float* __restrict__ part) {
  const float* Km = w + K_OFF;
  const float* v  = w + V_OFF;
  const int wave = threadIdx.x >> 5, lane = threadIdx.x & 31;
  const int i  = blockIdx.x * 8 + wave;
  const int j0 = blockIdx.y * 1024;
  float acc = 0.0f;
#pragma unroll 4
  for (int it = 0; it < 32; ++it) {
    int j = j0 + lane + it * 32;
    acc += Km[(size_t)i * NN + j] * v[j];
  }
  acc = wave_sum(acc);
  if (lane == 0) part[blockIdx.y * NN + i] = acc;
}
__global__ void k_ufin(float* __restrict__ w) {
  const float* part = w + PART_OFF;
  int i = blockIdx.x * blockDim.x + threadIdx.x;
  float s = 0.0f;
#pragma unroll
  for (int c = 0; c < 4; ++c) s += part[c * NN + i];
  w[U_OFF + i] = (1.0f / NN) / (s + FEPS);
}

// ---------------- ot = sum_ij u_i K_ij v_j C_ij ----------------
__global__ void __launch_bounds__(256) k_ot(float* __restrict__ w) {
  const float* S  = w + S_OFF;
  const float* Km = w + K_OFF;
  const int stride = gridDim.x * blockDim.x;
  float acc = 0.0f;
  for (size_t idx = blockIdx.x * blockDim.x + threadIdx.x; idx < (size_t)NN * NN; idx += stride) {
    int i = (int)(idx >> 12), j = (int)(idx & (NN - 1));
    float c = fmaxf(w[XSQ_OFF + i] + w[YSQ_OFF + j] - 2.0f * S[idx], 0.0f);
    acc += w[U_OFF + i] * Km[idx] * w[V_OFF + j] * c;
  }
  acc = wave_sum(acc);
  if ((threadIdx.x & 31) == 0) atomicAdd(w + ACC_OFF + 3, acc);
}

// ---------------- masked reconstruction loss pieces ----------------
__global__ void __launch_bounds__(256) k_msrp(const float* __restrict__ p, const float* __restrict__ t,
                                              const int* __restrict__ mask, float* __restrict__ w) {
  const int total = 256 * 128 * 64;
  const int stride = gridDim.x * blockDim.x;
  float sq = 0.0f, cnt = 0.0f;
  for (int idx = blockIdx.x * blockDim.x + threadIdx.x; idx < total; idx += stride) {
    float m = (float)mask[idx >> 6];
    float d = (p[idx] - t[idx]) * m;
    sq += d * d;
    if ((idx & 63) == 0) cnt += m;
  }
  sq = wave_sum(sq); cnt = wave_sum(cnt);
  if ((threadIdx.x & 31) == 0) { atomicAdd(w + ACC_OFF + 0, sq); atomicAdd(w + ACC_OFF + 1, cnt); }
}

// ---------------- physics loss pieces ----------------
__global__ void __launch_bounds__(256) k_phys(const float* __restrict__ sp, float* __restrict__ w) {
  const int L = 4096, total = 256 * 4096;
  const int stride = gridDim.x * blockDim.x;
  float smooth = 0.0f, nonneg = 0.0f, deriv = 0.0f, pknum = 0.0f, pkcnt = 0.0f;
  for (int idx = blockIdx.x * blockDim.x + threadIdx.x; idx < total; idx += stride) {
    int c = idx & (L - 1);
    float s0 = sp[idx];
    nonneg += fmaxf(-s0, 0.0f);
    if (c < L - 1) {
      float s1 = sp[idx + 1];
      float d = s1 - s0;
      smooth += d * d;
      if (c < L - 2) {
        float s2 = sp[idx + 2];
        if (s1 > s0 && s1 > s2) { float a = s2 - s0; pknum += a * a; pkcnt += 1.0f; }
        if (c < L - 3) {
          float s3 = sp[idx + 3];
          float d3 = s3 - 3.0f * s2 + 3.0f * s1 - s0;
          deriv += d3 * d3;
        }
      }
    }
  }
  smooth = wave_sum(smooth); nonneg = wave_sum(nonneg); deriv = wave_sum(deriv);
  pknum = wave_sum(pknum);   pkcnt = wave_sum(pkcnt);
  if ((threadIdx.x & 31) == 0) {
    atomicAdd(w + ACC_OFF + 4, smooth);
    atomicAdd(w + ACC_OFF + 5, nonneg);
    atomicAdd(w + ACC_OFF + 6, deriv);
    atomicAdd(w + ACC_OFF + 7, pknum);
    atomicAdd(w + ACC_OFF + 8, pkcnt);
  }
}

// ---------------- final combine ----------------
__global__ void k_final(const float* __restrict__ w, float* __restrict__ out) {
  const float* a = w + ACC_OFF;
  float msrp     = a[0] / (a[1] * 64.0f + FEPS);
  float contrast = a[2] / (float)NN;
  float ot       = a[3];
  float phys = 0.05f * a[4] / (256.0f * 4095.0f)
             + 0.02f * a[5] / (256.0f * 4096.0f)
             + 0.03f * a[6] / (256.0f * 4093.0f)
             + 0.02f * a[7] / (a[8] + FEPS);
  out[0] = 1.0f * msrp + 0.5f * contrast + 0.1f * ot + phys;
}

extern "C" void kernel_launch(void* const* d_in, const int* in_sizes, int n_in,
                              void* d_out, int out_size, void* d_ws, size_t ws_size,
                              hipStream_t stream) {
  const float* pred   = (const float*)d_in[0];
  const float* target = (const float*)d_in[1];
  const int*   mask   = (const int*)d_in[2];
  const float* z1     = (const float*)d_in[3];
  const float* z2     = (const float*)d_in[4];
  const float* spec   = (const float*)d_in[5];
  float* w   = (float*)d_ws;
  float* out = (float*)d_out;

  k_init<<<16, 256, 0, stream>>>(w);
  k_rownorms<<<1024, 256, 0, stream>>>(z1, z2, w);
  k_gemm<<<dim3(NN / 128, NN / 64), 256, 0, stream>>>(z1, z2, w + S_OFF);
  k_rows<<<NN, 256, 0, stream>>>(w);

  for (int it = 0; it < 100; ++it) {
    k_vpart<<<dim3(16, 16), 256, 0, stream>>>(w, w + PART_OFF);
    k_vfin<<<16, 256, 0, stream>>>(w);
    k_upart<<<dim3(512, 4), 256, 0, stream>>>(w, w + PART_OFF);
    k_ufin<<<16, 256, 0, stream>>>(w);
  }

  k_ot<<<4096, 256, 0, stream>>>(w);
  k_msrp<<<2048, 256, 0, stream>>>(pred, target, mask, w);
  k_phys<<<1024, 256, 0, stream>>>(spec, w);
  k_final<<<1, 1, 0, stream>>>(w, out);
}